// intra_topk_6107443494987
// MI455X (gfx1250) — compile-verified
//
#include <hip/hip_runtime.h>
#include <math.h>

typedef __attribute__((ext_vector_type(2))) float v2f;
typedef __attribute__((ext_vector_type(8))) float v8f;

#define SEG 16
#define NSEG 256
#define TOPK 51
#define LDS_STRIDE 260   // 256 + 4 pad: rows 8 apart land on different bank groups

// One block = one (b,d) pair x one 16-row slab of the 256x256 cosine matrix.
// 8 waves/block; each wave computes two 16x16 tiles with V_WMMA_F32_16X16X4_F32.
__global__ __launch_bounds__(256) void gram_topk_kernel(const float* __restrict__ x,
                                                        float* __restrict__ out) {
    __shared__ float slab[16 * LDS_STRIDE];

    const int blk   = blockIdx.x;
    const int pair  = blk >> 4;      // b*16 + d  (x is (32,16,4096))
    const int itile = blk & 15;      // which 16-row slab of the 256x256 matrix
    const int b     = pair >> 4;
    const int d     = pair & 15;

    const float* xp = x + (size_t)pair * 4096;

    const int lane  = threadIdx.x & 31;
    const int wave  = threadIdx.x >> 5;
    const int mrow  = lane & 15;       // M (or N) index held by this lane
    const int khalf = (lane >> 4) * 2; // K sub-offset: lanes 16-31 hold K+2,K+3

    // ---- A fragment: rows itile*16 .. itile*16+15, all 16 K values ----
    // ISA layout 16x4 f32 A: lane L holds (M=L&15, K = 2*(L>>4)+{0,1}) per chunk.
    v2f afrag[4];
    float rpart = 0.f;
#pragma unroll
    for (int kk = 0; kk < 4; ++kk) {
        const float* p = xp + (itile * 16 + mrow) * SEG + 4 * kk + khalf;
        v2f v; v.x = p[0]; v.y = p[1];
        afrag[kk] = v;
        rpart += v.x * v.x + v.y * v.y;
    }
    // fold the two half-wave K partials: lanes L and L^16 together cover K=0..15
    float rowsq = rpart + __shfl_xor(rpart, 16);   // rowsq of segment (L&15)

    const int n_col = lane & 15;
    const int hi    = lane >> 4;

    // ---- each wave computes column tiles j = 2*wave, 2*wave+1 ----
#pragma unroll
    for (int jj = 0; jj < 2; ++jj) {
        const int jtile = wave * 2 + jj;

        // B fragment (4x16 K x N): identical per-lane layout to A for a Gram matrix.
        v2f bfrag[4];
        float cpart = 0.f;
#pragma unroll
        for (int kk = 0; kk < 4; ++kk) {
            const float* p = xp + (jtile * 16 + mrow) * SEG + 4 * kk + khalf;
            v2f v; v.x = p[0]; v.y = p[1];
            bfrag[kk] = v;
            cpart += v.x * v.x + v.y * v.y;
        }
        float colsq = cpart + __shfl_xor(cpart, 16); // colsq of segment (L&15) == column N

        v8f c = {};
#pragma unroll
        for (int kk = 0; kk < 4; ++kk) {
            // D = A*B + C, fp32, 16x16x4
            c = __builtin_amdgcn_wmma_f32_16x16x4_f32(
                    false, afrag[kk], false, bfrag[kk],
                    (short)0, c, false, false);
        }

        // C/D layout: vgpr v, lane L -> (M = v + 8*(L>>4), N = L&15)
        const float cinv = rsqrtf(colsq);   // column norm already in the right lane
#pragma unroll
        for (int v = 0; v < 8; ++v) {
            const int M = v + 8 * hi;
            const float rq = __shfl(rowsq, M);        // broadcast row norm^2 from lane M
            const float val = c[v] * cinv * rsqrtf(rq);
            slab[M * LDS_STRIDE + jtile * 16 + n_col] = val;
        }
    }

    __syncthreads();

    // ---- per-row top-51 threshold + masked transposed write ----
    // 16 rows, 8 waves -> rows (wave) and (wave+8). 256 cols -> 8 per lane.
#pragma unroll
    for (int rr = 0; rr < 2; ++rr) {
        const int row = wave + rr * 8;

        float vals[8], work[8];
#pragma unroll
        for (int t = 0; t < 8; ++t) {
            vals[t] = slab[row * LDS_STRIDE + t * 32 + lane];
            work[t] = vals[t];
        }

        float thresh = -INFINITY;
        for (int it = 0; it < TOPK; ++it) {
            // local max over this lane's 8 survivors
            float m = -INFINITY; int slot = 0;
#pragma unroll
            for (int t = 0; t < 8; ++t)
                if (work[t] > m) { m = work[t]; slot = t; }
            // wave32 max reduce
            float wm = m;
#pragma unroll
            for (int off = 16; off > 0; off >>= 1) {
                float o = __shfl_xor(wm, off);
                wm = o > wm ? o : wm;
            }
            // lowest lane holding the max extracts it
            unsigned long long bal = __ballot(m == wm);
            int owner = __ffsll(bal) - 1;
            if (lane == owner) work[slot] = -INFINITY;
            thresh = wm;    // after 51 rounds: 51st-largest value
        }

        // output layout: (d, b, i, j)
        const size_t obase =
            (((size_t)(d * 32 + b) * 256) + (size_t)(itile * 16 + row)) * 256;
#pragma unroll
        for (int t = 0; t < 8; ++t) {
            const float v = vals[t];
            out[obase + t * 32 + lane] = (v >= thresh) ? v : 0.f;
        }
    }
}

extern "C" void kernel_launch(void* const* d_in, const int* in_sizes, int n_in,
                              void* d_out, int out_size, void* d_ws, size_t ws_size,
                              hipStream_t stream) {
    (void)in_sizes; (void)n_in; (void)out_size; (void)d_ws; (void)ws_size;
    const float* x = (const float*)d_in[0];
    float* out = (float*)d_out;
    // 512 (b,d) pairs x 16 row-slabs
    gram_topk_kernel<<<dim3(512 * 16), dim3(256), 0, stream>>>(x, out);
}